// SelfAttention2D_Block_F_69501160784162
// MI455X (gfx1250) — compile-verified
//
#include <hip/hip_runtime.h>

// Problem constants from the reference
#define N_DIM 64
#define T_DIM 96
#define F_DIM 32
#define ND    8
#define D_DIM 128

typedef float v2f __attribute__((ext_vector_type(2)));
typedef float v8f __attribute__((ext_vector_type(8)));

#define V8F_ZERO {0.f,0.f,0.f,0.f,0.f,0.f,0.f,0.f}

// D = A(16x4 f32) * B(4x16 f32) + C(16x16 f32), wave32 matrix op.
// A lane layout: lanes0-15 hold M=lane, K={0,1} in (a0,a1); lanes16-31 K={2,3}.
// B lane layout: lanes0-15 hold N=lane, K={0,1} in (b0,b1); lanes16-31 K={2,3}.
// C/D: VGPR r -> row r + 8*(lane>=16), col lane&15.
__device__ __forceinline__ v8f wmma_f32(v2f a, v2f b, v8f c) {
  return __builtin_amdgcn_wmma_f32_16x16x4_f32(
      /*neg_a=*/false, a, /*neg_b=*/false, b,
      /*c_mod=*/(short)0, c, /*reuse_a=*/false, /*reuse_b=*/false);
}

// ---------------------------------------------------------------------------
// Kernel 0: transpose w1 (8 x 32768) -> w1t (32768 x 8) so the fused epilogue
// in kernel 2 can fetch all 8 output coefficients with two b128 loads.
// grid = 128 x 256 threads, one thread per idx; reads coalesced across lanes.
// ---------------------------------------------------------------------------
__global__ __launch_bounds__(256) void w1t_kernel(
    const float* __restrict__ w1, float* __restrict__ w1t)
{
  const int idx = blockIdx.x * 256 + threadIdx.x;  // 0..32767
  float4 lo, hi;
  lo.x = w1[0 * 32768 + idx];
  lo.y = w1[1 * 32768 + idx];
  lo.z = w1[2 * 32768 + idx];
  lo.w = w1[3 * 32768 + idx];
  hi.x = w1[4 * 32768 + idx];
  hi.y = w1[5 * 32768 + idx];
  hi.z = w1[6 * 32768 + idx];
  hi.w = w1[7 * 32768 + idx];
  float4* o = (float4*)w1t;
  o[idx * 2]     = lo;
  o[idx * 2 + 1] = hi;
}

// ---------------------------------------------------------------------------
// Kernel 1: Q/K/V projections. q_[f,kh] = X_f(64x96) @ Wbase[kh](96x128) + bias
// grid = F*ND*3 blocks, 256 threads (8 waves); wave = N-tile; 4 M-tiles
// register-blocked so each B fragment is loaded once per K-step.
// ---------------------------------------------------------------------------
__global__ __launch_bounds__(256) void qkv_kernel(
    const float* __restrict__ x,
    const float* __restrict__ Wq, const float* __restrict__ Wk, const float* __restrict__ Wv,
    const float* __restrict__ bq, const float* __restrict__ bk, const float* __restrict__ bv,
    float* __restrict__ qout, float* __restrict__ kout, float* __restrict__ vout)
{
  const int bid   = blockIdx.x;       // 0..767
  const int which = bid % 3;
  const int fh    = bid / 3;          // f*ND + kh, 0..255
  const int f     = fh / ND;
  const int kh    = fh % ND;          // tiled weights: first ND slices are the base
  const float* W  = (which == 0) ? Wq : (which == 1) ? Wk : Wv;
  const float* BI = (which == 0) ? bq : (which == 1) ? bk : bv;
  float* OUT      = (which == 0) ? qout : (which == 1) ? kout : vout;

  const int lane = threadIdx.x & 31;
  const int nt   = threadIdx.x >> 5;  // 0..7
  const int l16  = lane & 15;
  const int hi   = lane >> 4;
  const int dcol = nt * 16 + l16;

  const float* Wh = W + kh * T_DIM * D_DIM;

  v8f acc[4] = {V8F_ZERO, V8F_ZERO, V8F_ZERO, V8F_ZERO};
  #pragma unroll 4
  for (int ks = 0; ks < 24; ++ks) {          // K = 96 (T dim)
    const int t0 = ks * 4 + 2 * hi;
    v2f b;
    b[0] = Wh[(t0) * D_DIM + dcol];
    b[1] = Wh[(t0 + 1) * D_DIM + dcol];
    #pragma unroll
    for (int mt = 0; mt < 4; ++mt) {
      const int m = mt * 16 + l16;
      v2f a;
      a[0] = x[(m * T_DIM + t0) * F_DIM + f];     // X_f[m, t] = x[m, t, f]
      a[1] = x[(m * T_DIM + t0 + 1) * F_DIM + f];
      acc[mt] = wmma_f32(a, b, acc[mt]);
    }
  }
  const float bias = BI[fh * D_DIM + dcol];
  #pragma unroll
  for (int mt = 0; mt < 4; ++mt) {
    #pragma unroll
    for (int r = 0; r < 8; ++r) {
      const int row = mt * 16 + r + 8 * hi;
      OUT[(fh * N_DIM + row) * D_DIM + dcol] = acc[mt][r] + bias;
    }
  }
}

// ---------------------------------------------------------------------------
// Kernel 2: fused alpha + w1 contraction (the dominant GEMM, 8.6 GFLOP).
// For each (f,g): h_pre[f,g,j] = sum_{kh,n,m} alpha[f,g,kh,n,m] * w1[j, kh*4096+n*64+m]
// Per wave: full 64x64 head tile as a 4x4 block of 16x16 WMMA tiles, K-outer
// with fragment reuse: 8 b64 loads -> 16 WMMAs per K-step.
// grid = 1024 blocks (f,g); 256 threads = 8 waves, wave = head.
// ---------------------------------------------------------------------------
__global__ __launch_bounds__(256) void alpha_h_kernel(
    const float* __restrict__ qbuf, const float* __restrict__ kbuf,
    const float* __restrict__ w1t, float* __restrict__ hpre)
{
  const int f    = blockIdx.x >> 5;
  const int g    = blockIdx.x & 31;
  const int lane = threadIdx.x & 31;
  const int kh   = threadIdx.x >> 5;   // 0..7
  const int l16  = lane & 15;
  const int hi   = lane >> 4;
  const float scale = 0.08838834764831845f;  // D^-0.5

  // v2f-granular views: element (row, d) -> v2f index row*64 + d/2
  const v2f* qv = (const v2f*)(qbuf + (f * ND + kh) * N_DIM * D_DIM);
  const v2f* kv = (const v2f*)(kbuf + (g * ND + kh) * N_DIM * D_DIM);

  __builtin_prefetch(w1t + kh * 4096 * 8, 0, 1);   // global_prefetch_b8

  v8f acc[4][4];
  #pragma unroll
  for (int mt = 0; mt < 4; ++mt)
    #pragma unroll
    for (int nt = 0; nt < 4; ++nt) acc[mt][nt] = (v8f)V8F_ZERO;

  #pragma unroll 2
  for (int ks = 0; ks < 32; ++ks) {        // K = 128 (D dim)
    const int p = ks * 2 + hi;             // v2f index of this lane's K-pair
    v2f afr[4], bfr[4];
    #pragma unroll
    for (int mt = 0; mt < 4; ++mt) afr[mt] = qv[(mt * 16 + l16) * 64 + p];
    #pragma unroll
    for (int nt = 0; nt < 4; ++nt) bfr[nt] = kv[(nt * 16 + l16) * 64 + p];
    #pragma unroll
    for (int mt = 0; mt < 4; ++mt)
      #pragma unroll
      for (int nt = 0; nt < 4; ++nt)
        acc[mt][nt] = wmma_f32(afr[mt], bfr[nt], acc[mt][nt]);
  }

  // Fused contraction with w1t (idx-major, 8 contiguous j): 2 b128 per element.
  float partial[8];
  #pragma unroll
  for (int j = 0; j < 8; ++j) partial[j] = 0.f;
  const float4* w1t4 = (const float4*)w1t;

  #pragma unroll
  for (int mt = 0; mt < 4; ++mt) {
    #pragma unroll
    for (int nt = 0; nt < 4; ++nt) {
      #pragma unroll
      for (int r = 0; r < 8; ++r) {
        const int n  = mt * 16 + r + 8 * hi;
        const int m  = nt * 16 + l16;
        const float av = acc[mt][nt][r];   // scale applied once at the end
        const int idx  = kh * 4096 + n * 64 + m;
        const float4 wa = w1t4[idx * 2];
        const float4 wb = w1t4[idx * 2 + 1];
        partial[0] += av * wa.x;  partial[1] += av * wa.y;
        partial[2] += av * wa.z;  partial[3] += av * wa.w;
        partial[4] += av * wb.x;  partial[5] += av * wb.y;
        partial[6] += av * wb.z;  partial[7] += av * wb.w;
      }
    }
  }

  __shared__ float red[256 * 8];
  #pragma unroll
  for (int j = 0; j < 8; ++j) red[threadIdx.x * 8 + j] = partial[j] * scale;
  __syncthreads();
  if (threadIdx.x < 8) {
    float s = 0.f;
    for (int l = 0; l < 256; ++l) s += red[l * 8 + threadIdx.x];
    hpre[(f * 32 + g) * 8 + threadIdx.x] = s;
  }
}

// ---------------------------------------------------------------------------
// Kernel 3: logits + softmax. One wave; thread = row f.
// ---------------------------------------------------------------------------
__global__ __launch_bounds__(32) void logits_softmax_kernel(
    const float* __restrict__ hpre, const float* __restrict__ b1,
    const float* __restrict__ w2, const float* __restrict__ b2,
    float* __restrict__ att_ws, float* __restrict__ att_out)
{
  const int f = threadIdx.x;  // 0..31
  float logit[32];
  float mx = -1e30f;
  #pragma unroll
  for (int g = 0; g < 32; ++g) {
    float s = b2[0];
    #pragma unroll
    for (int j = 0; j < 8; ++j) {
      float h = hpre[(f * 32 + g) * 8 + j] + b1[j];
      h = h > 0.f ? h : 0.f;
      s += h * w2[j];
    }
    logit[g] = s;
    mx = s > mx ? s : mx;
  }
  float sum = 0.f;
  #pragma unroll
  for (int g = 0; g < 32; ++g) { logit[g] = __expf(logit[g] - mx); sum += logit[g]; }
  const float inv = 1.f / sum;
  #pragma unroll
  for (int g = 0; g < 32; ++g) {
    const float a = logit[g] * inv;
    att_ws[f * 32 + g]  = a;
    att_out[f * 32 + g] = a;
  }
}

// ---------------------------------------------------------------------------
// Kernel 4: b = att @ v, stored directly in (F, N, D*ND) layout (j = d*ND + kh).
// float4-vectorized: each thread owns 4 consecutive d (b128 coalesced loads).
// grid = 2048 x 256.
// ---------------------------------------------------------------------------
__global__ __launch_bounds__(256) void bmix_kernel(
    const float* __restrict__ att, const float* __restrict__ vbuf,
    float* __restrict__ bmat)
{
  const int tid = blockIdx.x * 256 + threadIdx.x;  // 0 .. 524287
  const int d4 = tid & 31;           // d = 4*d4
  const int kh = (tid >> 5) & 7;
  const int n  = (tid >> 8) & 63;
  const int f  = tid >> 14;
  const float4* v4 = (const float4*)vbuf;
  float4 acc = {0.f, 0.f, 0.f, 0.f};
  #pragma unroll 8
  for (int g = 0; g < 32; ++g) {
    const float w = att[f * 32 + g];
    const float4 vv = v4[((g * ND + kh) * N_DIM + n) * 32 + d4];
    acc.x += w * vv.x;  acc.y += w * vv.y;
    acc.z += w * vv.z;  acc.w += w * vv.w;
  }
  const int base = ((f * N_DIM + n) * D_DIM + d4 * 4) * ND + kh;
  bmat[base]      = acc.x;
  bmat[base + 8]  = acc.y;
  bmat[base + 16] = acc.z;
  bmat[base + 24] = acc.w;
}

// ---------------------------------------------------------------------------
// Kernel 5: y = relu(bmat(2048x1024) @ w3^T + b3) -> (2048 x 128), WMMA.
// grid = 64 blocks, 2 M-tiles each (B fragment reused); 8 waves = 8 N-tiles.
// ---------------------------------------------------------------------------
__global__ __launch_bounds__(256) void y1_kernel(
    const float* __restrict__ bmat, const float* __restrict__ w3,
    const float* __restrict__ b3, float* __restrict__ ybuf)
{
  const int mt0  = blockIdx.x * 2;      // 0..126
  const int lane = threadIdx.x & 31;
  const int nt   = threadIdx.x >> 5;    // 0..7
  const int l16  = lane & 15;
  const int hi   = lane >> 4;
  const int ocol = nt * 16 + l16;

  v8f acc0 = V8F_ZERO, acc1 = V8F_ZERO;
  const v2f* a0 = (const v2f*)(bmat + (mt0 * 16 + l16) * 1024);
  const v2f* a1 = (const v2f*)(bmat + ((mt0 + 1) * 16 + l16) * 1024);
  const v2f* bv = (const v2f*)(w3 + ocol * 1024);   // B[k, o] = w3[o, k]
  #pragma unroll 4
  for (int ks = 0; ks < 256; ++ks) {                // K = 1024
    const int p = ks * 2 + hi;
    const v2f b = bv[p];
    acc0 = wmma_f32(a0[p], b, acc0);
    acc1 = wmma_f32(a1[p], b, acc1);
  }
  const float bias = b3[ocol];
  #pragma unroll
  for (int r = 0; r < 8; ++r) {
    const int row0 = mt0 * 16 + r + 8 * hi;
    float y0 = acc0[r] + bias;
    float y1 = acc1[r] + bias;
    ybuf[row0 * 128 + ocol]        = y0 > 0.f ? y0 : 0.f;
    ybuf[(row0 + 16) * 128 + ocol] = y1 > 0.f ? y1 : 0.f;
  }
}

// ---------------------------------------------------------------------------
// Kernel 6: out = y(2048x128) @ w4^T + b4, scatter-stored as (N, T, F).
// grid = 64 blocks, 2 M-tiles each; 6 waves = 6 N-tiles (T = 96).
// ---------------------------------------------------------------------------
__global__ __launch_bounds__(192) void y2_kernel(
    const float* __restrict__ ybuf, const float* __restrict__ w4,
    const float* __restrict__ b4, float* __restrict__ out)
{
  const int mt0  = blockIdx.x * 2;      // 0..126
  const int lane = threadIdx.x & 31;
  const int nt   = threadIdx.x >> 5;    // 0..5
  const int l16  = lane & 15;
  const int hi   = lane >> 4;
  const int tcol = nt * 16 + l16;

  v8f acc0 = V8F_ZERO, acc1 = V8F_ZERO;
  const v2f* a0 = (const v2f*)(ybuf + (mt0 * 16 + l16) * 128);
  const v2f* a1 = (const v2f*)(ybuf + ((mt0 + 1) * 16 + l16) * 128);
  const v2f* bv = (const v2f*)(w4 + tcol * 128);    // B[o, t] = w4[t, o]
  #pragma unroll 4
  for (int ks = 0; ks < 32; ++ks) {                 // K = 128
    const int p = ks * 2 + hi;
    const v2f b = bv[p];
    acc0 = wmma_f32(a0[p], b, acc0);
    acc1 = wmma_f32(a1[p], b, acc1);
  }
  const float bias = b4[tcol];
  #pragma unroll
  for (int r = 0; r < 8; ++r) {
    const int row0 = mt0 * 16 + r + 8 * hi;         // row = f*64 + n
    const int f0 = row0 >> 6,        n0 = row0 & 63;
    const int f1 = (row0 + 16) >> 6, n1 = (row0 + 16) & 63;
    out[(n0 * T_DIM + tcol) * F_DIM + f0] = acc0[r] + bias;
    out[(n1 * T_DIM + tcol) * F_DIM + f1] = acc1[r] + bias;
  }
}

// ---------------------------------------------------------------------------
extern "C" void kernel_launch(void* const* d_in, const int* in_sizes, int n_in,
                              void* d_out, int out_size, void* d_ws, size_t ws_size,
                              hipStream_t stream) {
  const float* x  = (const float*)d_in[0];
  const float* Wq = (const float*)d_in[1];
  const float* Wk = (const float*)d_in[2];
  const float* Wv = (const float*)d_in[3];
  const float* bq = (const float*)d_in[4];
  const float* bk = (const float*)d_in[5];
  const float* bv = (const float*)d_in[6];
  const float* w1 = (const float*)d_in[7];
  const float* b1 = (const float*)d_in[8];
  const float* w2 = (const float*)d_in[9];
  const float* b2 = (const float*)d_in[10];
  const float* w3 = (const float*)d_in[11];
  const float* b3 = (const float*)d_in[12];
  const float* w4 = (const float*)d_in[13];
  const float* b4 = (const float*)d_in[14];

  float* out = (float*)d_out;               // [0,196608) = (N,T,F); [196608,197632) = att
  float* ws  = (float*)d_ws;

  // Workspace layout (floats), with overlays (q dead after k2, k dead after k2):
  float* qbuf = ws;                 // 2,097,152  (reused as bmat by k4/k5)
  float* kbuf = ws + 2097152;       // 2,097,152  (reused as y by k5/k6)
  float* vbuf = ws + 4194304;       // 2,097,152
  float* hpre = ws + 6291456;       //     8,192
  float* attw = ws + 6299648;       //     1,024
  float* w1t  = ws + 6300672;       //   262,144 (16B aligned)

  w1t_kernel<<<128, 256, 0, stream>>>(w1, w1t);
  qkv_kernel<<<768, 256, 0, stream>>>(x, Wq, Wk, Wv, bq, bk, bv, qbuf, kbuf, vbuf);
  alpha_h_kernel<<<1024, 256, 0, stream>>>(qbuf, kbuf, w1t, hpre);
  logits_softmax_kernel<<<1, 32, 0, stream>>>(hpre, b1, w2, b2, attw, out + 196608);
  bmix_kernel<<<2048, 256, 0, stream>>>(attw, vbuf, qbuf /*bmat*/);
  y1_kernel<<<64, 256, 0, stream>>>(qbuf /*bmat*/, w3, b3, kbuf /*y*/);
  y2_kernel<<<64, 192, 0, stream>>>(kbuf /*y*/, w4, b4, out);
}